// YOLOv1Loss_55164559950272
// MI455X (gfx1250) — compile-verified
//
#include <hip/hip_runtime.h>

#define CH        30
#define TILE      256
#define CELLS     (16384 * 7 * 7)      // 802816 = 3136 * 256
#define NTILES    (CELLS / TILE)       // 3136
#define TILE_F    (TILE * CH)          // 7680 floats per array per tile
#define TILE_F4   (TILE_F / 4)         // 1920 float4 per array per tile

typedef float v2f __attribute__((ext_vector_type(2)));
typedef float v8f __attribute__((ext_vector_type(8)));

// Full wave32 sum via V_WMMA_F32_16X16X4_F32.
// A (16x4): lane's partial in VGPR0, 0 in VGPR1 -> A[m][0]=v[m], A[m][2]=v[16+m].
// B (4x16) = ones. D[m][n] = v[m] + v[m+16] = r[m]; lane (n,half h) holds
// D[8h+j][n]=r[8h+j] in VGPR j. Summing the 8 D regs gives half-sums; one
// xor-16 shuffle completes the 32-lane total (broadcast to all lanes).
// Requires EXEC all-ones (call only from fully convergent code).
__device__ __forceinline__ float wave_reduce_sum(float v) {
    v2f a; a[0] = v;    a[1] = 0.0f;
    v2f b; b[0] = 1.0f; b[1] = 1.0f;
    v8f c = {};
    v8f d = __builtin_amdgcn_wmma_f32_16x16x4_f32(
        /*neg_a=*/false, a, /*neg_b=*/false, b,
        /*c_mod=*/(short)0, c, /*reuse_a=*/false, /*reuse_b=*/false);
    float s = ((d[0] + d[1]) + (d[2] + d[3])) + ((d[4] + d[5]) + (d[6] + d[7]));
    s += __shfl_xor(s, 16, 32);
    return s;
}

__device__ __forceinline__ float iou_f(float b0, float b1, float b2, float b3,
                                       float l0, float l1, float l2, float l3) {
    float area1 = b2 * b3;
    float area2 = l2 * l3;
    float maxL = fmaxf(b0 - 0.5f * b2, l0 - 0.5f * l2);
    float minR = fminf(b0 + 0.5f * b2, l0 + 0.5f * l2);
    float maxT = fmaxf(b1 - 0.5f * b3, l1 - 0.5f * l3);
    float minB = fminf(b1 + 0.5f * b3, l1 + 0.5f * l3);
    float iw = minR - maxL;
    float ih = minB - maxT;
    float inter = iw * ih;
    float iou = inter / (area1 + area2 - inter);
    return (iw > 0.0f && ih > 0.0f) ? iou : 0.0f;
}

__global__ void __launch_bounds__(TILE)
yolo_loss_partial(const float* __restrict__ preds,
                  const float* __restrict__ labels,
                  float* __restrict__ partials) {
    __shared__ float sp[TILE_F];
    __shared__ float sl[TILE_F];
    __shared__ float wsum[TILE / 32];

    const int tid = threadIdx.x;
    const size_t tileBase4 = (size_t)blockIdx.x * TILE_F4;

    // Stage 256 cells of both tensors into LDS via the CDNA5 async
    // Global->LDS DMA path (no VGPR round-trip, tracked by ASYNCcnt).
    // SADDR form: uniform 64-bit base in SGPRs, per-lane byte offset in VGPR,
    // LDS destination byte offset in VGPR (low 32 bits of flat shared addr).
    const float4* gp = (const float4*)preds + tileBase4;
    const float4* gl = (const float4*)labels + tileBase4;
    float4* sp4 = (float4*)sp;
    float4* sl4 = (float4*)sl;
#pragma unroll
    for (int i = tid; i < TILE_F4; i += TILE) {
        unsigned int dstP = (unsigned int)(uintptr_t)(sp4 + i);
        unsigned int dstL = (unsigned int)(uintptr_t)(sl4 + i);
        unsigned int off  = (unsigned int)i * 16u;
        asm volatile("global_load_async_to_lds_b128 %0, %1, %2"
                     :: "v"(dstP), "v"(off), "s"(gp) : "memory");
        asm volatile("global_load_async_to_lds_b128 %0, %1, %2"
                     :: "v"(dstL), "v"(off), "s"(gl) : "memory");
    }
    asm volatile("s_wait_asynccnt 0x0" ::: "memory");
    __syncthreads();

    // One cell per thread; stride-30 LDS reads are bank-conflict-free (wave32, 64 banks).
    const float* P = &sp[tid * CH];
    const float* L = &sl[tid * CH];

    float l0 = L[0], l1 = L[1], l2 = L[2], l3 = L[3], l4 = L[4];
    float p0 = P[0], p1 = P[1], p2 = P[2], p3 = P[3], p4 = P[4];
    float p5 = P[5], p6 = P[6], p7 = P[7], p8 = P[8], p9 = P[9];

    float iou1 = iou_f(p0, p1, p2, p3, l0, l1, l2, l3);
    float iou2 = iou_f(p5, p6, p7, p8, l0, l1, l2, l3);
    bool sel1 = iou1 > iou2;

    float contrib;
    if (l4 == 1.0f) {
        float xy, wh, o, no;
        if (sel1) {
            float dx = l0 - p0, dy = l1 - p1;
            xy = dx * dx + dy * dy;
            float dw = sqrtf(l2) - sqrtf(p2), dh = sqrtf(l3) - sqrtf(p3);
            wh = dw * dw + dh * dh;
            float od = iou1 - p4;
            o = od * od;
            no = p9 * p9;
        } else {
            float l5 = L[5], l6 = L[6], l7 = L[7], l8 = L[8];
            float dx = l5 - p5, dy = l6 - p6;
            xy = dx * dx + dy * dy;
            float dw = sqrtf(l7) - sqrtf(p7), dh = sqrtf(l8) - sqrtf(p8);
            wh = dw * dw + dh * dh;
            float od = iou2 - p9;
            o = od * od;
            no = p4 * p4;
        }
        float cls = 0.0f;
#pragma unroll
        for (int k = 10; k < CH; ++k) {
            float dc = L[k] - P[k];
            cls += dc * dc;
        }
        contrib = 5.0f * xy + wh + o + 0.5f * no + cls;
    } else {
        contrib = 0.5f * (p4 * p4 + p9 * p9);
    }

    // Fully convergent from here (no early exits): EXEC all ones for WMMA.
    float s = wave_reduce_sum(contrib);
    if ((tid & 31) == 0) wsum[tid >> 5] = s;
    __syncthreads();
    if (tid == 0) {
        float t = 0.0f;
#pragma unroll
        for (int j = 0; j < TILE / 32; ++j) t += wsum[j];
        partials[blockIdx.x] = t;
    }
}

__global__ void __launch_bounds__(TILE)
yolo_loss_finalize(const float* __restrict__ partials, float* __restrict__ out) {
    __shared__ float wsum[TILE / 32];
    float acc = 0.0f;
    for (int i = threadIdx.x; i < NTILES; i += TILE) acc += partials[i];
    float s = wave_reduce_sum(acc);
    if ((threadIdx.x & 31) == 0) wsum[threadIdx.x >> 5] = s;
    __syncthreads();
    if (threadIdx.x == 0) {
        float t = 0.0f;
#pragma unroll
        for (int j = 0; j < TILE / 32; ++j) t += wsum[j];
        out[0] = t * (1.0f / 16384.0f);
    }
}

extern "C" void kernel_launch(void* const* d_in, const int* in_sizes, int n_in,
                              void* d_out, int out_size, void* d_ws, size_t ws_size,
                              hipStream_t stream) {
    const float* preds  = (const float*)d_in[0];
    const float* labels = (const float*)d_in[1];
    float* out  = (float*)d_out;
    float* part = (float*)d_ws;   // NTILES * 4 = 12544 bytes of scratch

    yolo_loss_partial<<<NTILES, TILE, 0, stream>>>(preds, labels, part);
    yolo_loss_finalize<<<1, TILE, 0, stream>>>(part, out);
}